// SimKernel_27462020891153
// MI455X (gfx1250) — compile-verified
//
#include <hip/hip_runtime.h>

typedef __attribute__((ext_vector_type(2))) float v2f;
typedef __attribute__((ext_vector_type(8))) float v8f;

#define NPTS 12288
#define DIM  512
#define KNB  10
#define TILE 16
#define WAVES 8
#define COL_TILES (NPTS / TILE)            // 768
#define TILES_PER_WAVE (COL_TILES / WAVES) // 96
#define NEG_INF (-3.0e38f)

// ---------------- kernel 1: row squared norms -> d_ws ----------------
__global__ void row_norms_kernel(const float* __restrict__ F, float* __restrict__ sq) {
    const int wave = threadIdx.x >> 5;
    const int lane = threadIdx.x & 31;
    const int row  = blockIdx.x * 4 + wave;      // blockDim = 128 (4 waves)
    if (row >= NPTS) return;
    const float* p = F + (size_t)row * DIM;
    float s = 0.f;
    for (int k = lane; k < DIM; k += 32) { float x = p[k]; s += x * x; }
    #pragma unroll
    for (int off = 16; off > 0; off >>= 1) s += __shfl_xor(s, off, 32);
    if (lane == 0) sq[row] = s;
}

// ---------------- kernel 2: zero-fill adjacency (streaming b128 stores) ----------------
__global__ void zero_kernel(float4* __restrict__ out, size_t n4) {
    size_t i = (size_t)blockIdx.x * blockDim.x + threadIdx.x;
    const size_t stride = (size_t)gridDim.x * blockDim.x;
    const float4 z = make_float4(0.f, 0.f, 0.f, 0.f);
    for (; i < n4; i += stride) out[i] = z;
}

// ---------------- kernel 3: fused f32-WMMA GEMM + per-row top-10 + scatter ----------------
__launch_bounds__(256, 1)
__global__ void knn_adj_kernel(const float* __restrict__ F,
                               const float* __restrict__ sq,
                               float* __restrict__ adj) {
    __shared__ float Alds[TILE * DIM];                  // 32 KB: this block's 16-row A strip
    __shared__ float S[WAVES][TILE][TILE];              //  8 KB: per-wave score tile
    __shared__ float topS[WAVES][TILE][KNB];            //  5 KB: per-wave per-row top-10 scores
    __shared__ int   topC[WAVES][TILE][KNB];            //  5 KB: per-wave per-row top-10 cols

    const int tid  = threadIdx.x;
    const int wave = tid >> 5;
    const int lane = tid & 31;
    const int m    = lane & 15;            // A row / B col / C col within tile
    const int koff = (lane >> 4) << 1;     // lanes 16-31 carry K+2,K+3 halves
    const int I    = blockIdx.x * TILE;    // global row base for this block

    // Stage A strip (16 x 512 f32) into LDS cooperatively (b128 loads).
    {
        const float4* src = (const float4*)(F + (size_t)I * DIM);
        float4* dst = (float4*)Alds;
        for (int i = tid; i < TILE * DIM / 4; i += 256) dst[i] = src[i];
    }
    if (lane < TILE) {
        #pragma unroll
        for (int j = 0; j < KNB; ++j) { topS[wave][lane][j] = NEG_INF; topC[wave][lane][j] = 0; }
    }
    __syncthreads();

    for (int t = 0; t < TILES_PER_WAVE; ++t) {
        const int J = (t * WAVES + wave) * TILE;   // this wave's column tile

        // --- 16x16 f32 Gram tile: 128 x V_WMMA_F32_16X16X4_F32, K = 512 ---
        v8f c = {0.f, 0.f, 0.f, 0.f, 0.f, 0.f, 0.f, 0.f};
        const float* arow = &Alds[m * DIM + koff];                 // LDS  (ds_load_b64)
        const float* brow = &F[(size_t)(J + m) * DIM + koff];      // L2   (global_load_b64)
        #pragma unroll 4
        for (int k = 0; k < DIM; k += 4) {
            v2f a = *(const v2f*)(arow + k);
            v2f b = *(const v2f*)(brow + k);
            c = __builtin_amdgcn_wmma_f32_16x16x4_f32(
                    false, a, false, b, (short)0, c, false, false);
        }

        // --- epilogue in registers: score = 2*G - ||x_j||^2, mask diagonal ---
        const int col   = J + m;
        const float sqc = sq[col];
        const int rbase = (lane >> 4) * 8;   // C layout: VGPR v -> row v (lanes 0-15) / v+8 (lanes 16-31)
        #pragma unroll
        for (int v = 0; v < 8; ++v) {
            const int r = rbase + v;
            float s = 2.0f * c[v] - sqc;
            if (I + r == col) s = NEG_INF;
            S[wave][r][m] = s;
        }
        __syncthreads();

        // --- per-row top-10 maintenance: lanes 0..15 each own one row ---
        if (lane < TILE) {
            const int r = lane;
            float thresh = topS[wave][r][KNB - 1];
            for (int nn = 0; nn < TILE; ++nn) {
                const float s = S[wave][r][nn];
                if (s > thresh) {
                    int p = KNB - 1;
                    while (p > 0 && topS[wave][r][p - 1] < s) {
                        topS[wave][r][p] = topS[wave][r][p - 1];
                        topC[wave][r][p] = topC[wave][r][p - 1];
                        --p;
                    }
                    topS[wave][r][p] = s;
                    topC[wave][r][p] = J + nn;
                    thresh = topS[wave][r][KNB - 1];
                }
            }
        }
        __syncthreads();
    }

    // --- merge the 8 per-wave lists and scatter ones ---
    if (wave == 0 && lane < TILE) {
        const int r = lane;
        float bs[KNB]; int bc[KNB];
        #pragma unroll
        for (int j = 0; j < KNB; ++j) { bs[j] = NEG_INF; bc[j] = 0; }
        for (int w = 0; w < WAVES; ++w) {
            for (int j = 0; j < KNB; ++j) {
                const float s = topS[w][r][j];
                if (s > bs[KNB - 1]) {
                    int p = KNB - 1;
                    while (p > 0 && bs[p - 1] < s) { bs[p] = bs[p - 1]; bc[p] = bc[p - 1]; --p; }
                    bs[p] = s; bc[p] = topC[w][r][j];
                }
            }
        }
        float* outrow = adj + (size_t)(I + r) * NPTS;
        #pragma unroll
        for (int j = 0; j < KNB; ++j) outrow[bc[j]] = 1.0f;
    }
}

extern "C" void kernel_launch(void* const* d_in, const int* in_sizes, int n_in,
                              void* d_out, int out_size, void* d_ws, size_t ws_size,
                              hipStream_t stream) {
    (void)in_sizes; (void)n_in; (void)out_size; (void)ws_size;
    const float* F  = (const float*)d_in[0];
    float* adj      = (float*)d_out;
    float* sq       = (float*)d_ws;          // 12288 floats = 48 KB of scratch

    row_norms_kernel<<<NPTS / 4, 128, 0, stream>>>(F, sq);
    zero_kernel<<<2048, 256, 0, stream>>>((float4*)d_out, (size_t)NPTS * NPTS / 4);
    knn_adj_kernel<<<NPTS / TILE, 256, 0, stream>>>(F, sq, adj);
}